// EfficientAffineConvolution_19181323944311
// MI455X (gfx1250) — compile-verified
//
#include <hip/hip_runtime.h>

// EfficientAffineConvolution (bug-faithful torch-unfold view) for MI455X/gfx1250.
//
// out[b,c,n] = sum_{q=0..24} P[25n+q] * Kflat[25n+q]   per image (b,c)
//   f = 25n+q;  kk = f>>14; hw = f&16383; i=kk/5; j=kk%5
//   P[f] = latent[hw + (i-2)*128 + (j-2)]  with 2D zero-padding bounds on
//          row (hw>>7)+i-2 and col (hw&127)+j-2.
//
// Roofline: 227 MB traffic / 105 MFLOP = 0.46 FLOP/B -> HBM-bound, ~9.7us at
// 23.3 TB/s. Kernels stream (210 MB) moved by the Tensor Data Mover (one
// tensor_load_to_lds per 51.2 KB chunk, TENSORcnt, NT policy -> latent stays
// L2-resident). Latent reads use a buffer V# whose OOB-returns-zero semantics
// implement the row/linear padding for free (no med3 clamp, no row compare);
// per-q byte offsets fold into the buffer immediate offset.

#define HW_            16384
#define W_             128
#define KK_            25
#define CH_            512                 // outputs per workgroup
#define THREADS_       256
#define CHUNKS_PER_IMG 32                  // HW_/CH_
#define N_IMG          128                 // B*C = 8*16
#define KCHUNK_FLOATS  (CH_ * KK_)         // 12800 floats = 51,200 B LDS

typedef unsigned int u32;
typedef u32 v4u __attribute__((ext_vector_type(4)));
typedef int  v4i __attribute__((ext_vector_type(4)));
typedef int  v8i __attribute__((ext_vector_type(8)));

#if __has_builtin(__builtin_amdgcn_make_buffer_rsrc) && \
    __has_builtin(__builtin_amdgcn_raw_buffer_load_b32)
#define EAC_HAVE_BUFFER 1
#endif

// One-shot 1-D TDM DMA: `elems` f32 from global `gaddr` -> LDS byte addr.
// D# packing per CDNA5 ISA (groups 0/1; 2/3 unused).
__device__ __forceinline__ void tdm_load_1d(unsigned lds_addr, const void* gaddr,
                                            unsigned elems)
{
    const unsigned long long ga = (unsigned long long)(uintptr_t)gaddr;
    v4u g0;
    g0.x = 0x1u;                                    // count=1, no gather
    g0.y = lds_addr;                                // LDS byte address
    g0.z = (u32)ga;                                 // global_addr[31:0]
    g0.w = (u32)(ga >> 32) | 0x80000000u;           // global_addr[56:32] | type=2
    v8i g1;
    g1[0] = 0x00020000;                             // data_size=2 (4 B)
    g1[1] = (int)((elems & 0xFFFFu) << 16);         // tensor_dim0[15:0]
    g1[2] = (int)(((elems >> 16) & 0xFFFFu)         // tensor_dim0[31:16]
                  | (1u << 16));                    // tensor_dim1 = 1
    g1[3] = (int)((elems & 0xFFFFu) << 16);         // tile_dim0 (<= 65535)
    g1[4] = 1;                                      // tile_dim1=1, tile_dim2=0
    g1[5] = (int)elems;                             // tensor_dim0_stride[31:0]
    g1[6] = 0;                                      // stride hi / dim1_stride lo
    g1[7] = 0;                                      // dim1_stride hi

#if __has_builtin(__builtin_amdgcn_tensor_load_to_lds)
    const v4i z4 = (v4i)0;
  #if __clang_major__ >= 23
    const v8i z8 = (v8i)0;
    __builtin_amdgcn_tensor_load_to_lds(g0, g1, z4, z4, z8, /*cpol: TH_NT*/1);
  #else
    __builtin_amdgcn_tensor_load_to_lds(g0, g1, z4, z4, /*cpol: TH_NT*/1);
  #endif
#else
    asm volatile("tensor_load_to_lds %0, %1 th:TH_LOAD_NT"
                 :: "s"(g0), "s"(g1) : "memory");
#endif
}

__global__ __launch_bounds__(THREADS_)
void eac_kernel(const float* __restrict__ kernels,
                const float* __restrict__ latent,
                float* __restrict__ out)
{
    __shared__ float kbuf[KCHUNK_FLOATS];

    const int wg    = blockIdx.x;
    const int img   = wg >> 5;             // / CHUNKS_PER_IMG
    const int chunk = wg & (CHUNKS_PER_IMG - 1);
    const int n0    = chunk * CH_;
    const int tid   = threadIdx.x;

    const float* ksrc = kernels + (size_t)img * (HW_ * KK_) + (size_t)n0 * KK_;
    const float* lat  = latent  + (size_t)img * HW_;

    // ---- TDM: DMA this chunk's 51.2 KB of kernel weights into LDS ----
    // Issued by wave 0 only (tensor ops ignore EXEC; more waves would
    // replicate the DMA). TENSORcnt is per-wave -> wave 0 waits, barrier
    // releases the rest.
    if (tid < 32) {
        tdm_load_1d((unsigned)(uintptr_t)&kbuf[0], ksrc, KCHUNK_FLOATS);
        __builtin_amdgcn_s_wait_tensorcnt(0);
    }
    __syncthreads();

#ifdef EAC_HAVE_BUFFER
    // V# over this image's latent slice: raw-buffer OOB loads return 0, which
    // is exactly the zero-padding for row-OOB / linear-OOB accesses.
    __amdgpu_buffer_rsrc_t lrsrc = __builtin_amdgcn_make_buffer_rsrc(
        const_cast<float*>(lat), /*stride*/(short)0,
        /*num_records(bytes)*/HW_ * 4, /*flags*/0);
#endif

    // ---- each thread computes 2 outputs ----
    #pragma unroll
    for (int half = 0; half < 2; ++half) {
        const int ln = tid + half * THREADS_;      // local output 0..511
        const int n  = n0 + ln;
        const int t0 = n * KK_;
        const int kkA = t0 >> 14;                  // 0..24
        const int hw0 = t0 & (HW_ - 1);
        const int qb  = HW_ - hw0;                 // first q belonging to kk+1 (if <25)
        const int iA  = (kkA * 13) >> 6;           // kkA/5, exact for 0..24
        const int jA  = kkA - iA * 5;
        const int jA2 = jA - 2;
        const int offA  = (iA - 2) * W_ + jA2;
        const int lbase = ln * KK_;
#ifdef EAC_HAVE_BUFFER
        const int vbA = (hw0 + offA) * 4;          // loop-invariant voffset base
#endif

        float acc = 0.0f;
        #pragma unroll
        for (int q = 0; q < KK_; ++q) {
            const int c  = (hw0 + q) & (W_ - 1);
            const int ok = (int)(q < qb)
                         & (int)((unsigned)(c + jA2) < (unsigned)W_);
#ifdef EAC_HAVE_BUFFER
            // row-OOB and negative idx -> buffer OOB -> 0; 4q folds into imm.
            const float lv = __builtin_bit_cast(float,
                (u32)__builtin_amdgcn_raw_buffer_load_b32(lrsrc, vbA + q * 4, 0, 0));
            acc = fmaf(ok ? lv : 0.0f, kbuf[lbase + q], acc);
#else
            const int hwq = hw0 + q;
            const int r   = hwq >> 7;
            const int ok2 = ok & (int)((unsigned)(r + iA - 2) < (unsigned)W_);
            int idx = hwq + offA;
            idx = idx < 0 ? 0 : (idx > HW_ - 1 ? HW_ - 1 : idx);
            acc = fmaf(ok2 ? lat[idx] : 0.0f, kbuf[lbase + q], acc);
#endif
        }

        // Rare fixup: 25-segment straddles a kk boundary (~24 outputs / image).
        if (qb < KK_) {
            const int kkB  = kkA + 1;              // <= 24 whenever split occurs
            const int iB   = (kkB * 13) >> 6;
            const int jB   = kkB - iB * 5;
            const int jB2  = jB - 2;
            const int offB = (iB - 2) * W_ + jB2;
            for (int q = qb; q < KK_; ++q) {
                const int hwq = hw0 + q - HW_;     // wrapped hw
                const int c   = hwq & (W_ - 1);
                const int cok = (int)((unsigned)(c + jB2) < (unsigned)W_);
#ifdef EAC_HAVE_BUFFER
                const float lv = __builtin_bit_cast(float,
                    (u32)__builtin_amdgcn_raw_buffer_load_b32(
                        lrsrc, (hwq + offB) * 4, 0, 0));
                acc = fmaf(cok ? lv : 0.0f, kbuf[lbase + q], acc);
#else
                const int r   = hwq >> 7;
                const int ok2 = cok & (int)((unsigned)(r + iB - 2) < (unsigned)W_);
                int idx = hwq + offB;
                idx = idx < 0 ? 0 : (idx > HW_ - 1 ? HW_ - 1 : idx);
                acc = fmaf(ok2 ? lat[idx] : 0.0f, kbuf[lbase + q], acc);
#endif
            }
        }

        out[(size_t)img * HW_ + n] = acc;          // coalesced b32 store
    }
}

extern "C" void kernel_launch(void* const* d_in, const int* in_sizes, int n_in,
                              void* d_out, int out_size, void* d_ws, size_t ws_size,
                              hipStream_t stream) {
    const float* kernels = (const float*)d_in[0];  // (8,16,128,128,5,5) f32
    const float* latent  = (const float*)d_in[1];  // (8,16,128,128)     f32
    float* out = (float*)d_out;                    // (8,16,128,128)     f32

    eac_kernel<<<dim3(N_IMG * CHUNKS_PER_IMG), dim3(THREADS_), 0, stream>>>(
        kernels, latent, out);
}